// MDCA_87454124081405
// MI455X (gfx1250) — compile-verified
//
#include <hip/hip_runtime.h>
#include <hip/hip_bf16.h>

typedef float v2f __attribute__((ext_vector_type(2)));
typedef float v4f __attribute__((ext_vector_type(4)));
typedef float v8f __attribute__((ext_vector_type(8)));

#define B_SZ 128
#define C_SZ 1000
#define HW 196
#define CAM_ELEMS (B_SZ * C_SZ * HW)   // 25,088,000
#define CT 16                          // classes per tile (WMMA N)
#define NCT 63                         // ceil(1000/16)
#define BG 16                          // batches per block
#define NBG 8                          // 128/16
#define WAVES 4
#define BPW 4                          // batches per wave
#define TILE_FLOATS (CT * HW)          // 3136 floats = 12544 B per wave buffer
#define TILE_V4 (TILE_FLOATS / 4)      // 784 b128 transfers

#if __has_builtin(__builtin_amdgcn_wmma_f32_16x16x4_f32)
#define HAVE_WMMA_F32 1
#else
#define HAVE_WMMA_F32 0
#endif

typedef __attribute__((address_space(3))) void* as3v;

// Async global->LDS copy, 16 B per lane, tracked by ASYNCcnt.
// ISA: GLOBAL_LOAD_ASYNC_TO_LDS_B128: VDST = LDS byte-offset VGPR,
// VADDR = 64-bit global address pair, SADDR = off.
__device__ __forceinline__ void copy16_g2lds(const float* g, float* l) {
  unsigned lofs = (unsigned)(unsigned long long)(as3v)(void*)l;  // AS3 offset
  unsigned long long ga = (unsigned long long)g;
  asm volatile("global_load_async_to_lds_b128 %0, %1, off"
               :: "v"(lofs), "v"(ga) : "memory");
}

__device__ __forceinline__ void wait_async0() {
#if __has_builtin(__builtin_amdgcn_s_wait_asynccnt)
  __builtin_amdgcn_s_wait_asynccnt(0);
#else
  asm volatile("s_wait_asynccnt 0" ::: "memory");
#endif
}

// ---- kernel 0: zero the per-class accumulator workspace -------------------
__global__ void mdca_zero(float* __restrict__ p, int n) {
  int i = blockIdx.x * blockDim.x + threadIdx.x;
  if (i < n) p[i] = 0.0f;
}

// ---- kernel 1: streaming per-class reduction via async-LDS + WMMA ---------
// grid.x = 4 cams * 63 class-tiles * 8 batch-groups; block = 128 (4 waves)
__global__ void mdca_reduce(const float* __restrict__ cam0,
                            const float* __restrict__ cam1,
                            const float* __restrict__ cam2,
                            const float* __restrict__ cam3,
                            float* __restrict__ sums) {
  extern __shared__ float smem[];  // WAVES * TILE_FLOATS floats (50,176 B)

  const int bid   = blockIdx.x;
  const int cam   = bid / (NCT * NBG);
  const int rem   = bid % (NCT * NBG);
  const int ctile = rem / NBG;
  const int bg    = rem % NBG;
  const float* src = (cam == 0) ? cam0 : (cam == 1) ? cam1 : (cam == 2) ? cam2 : cam3;

  const int tid  = threadIdx.x;
  const int lane = tid & 31;
  const int wave = tid >> 5;
  const int c0   = ctile * CT;

  float* lbuf = smem + wave * TILE_FLOATS;  // private staging buffer per wave

  // B-operand read pattern: lane n (n<16) supplies K={0,1} of class column n,
  // lane 16+n supplies K={2,3}. With A == all-ones the WMMA result is
  // D[m][n] = sum_k B[k][n] regardless of the exact K-slot permutation.
  const int   ncls  = lane & 15;
  const int   koff  = (lane >> 4) << 1;  // 0 or 2
  const float* bbase = lbuf + ncls * HW + koff;

  v8f acc = {};            // 16x16 f32 accumulator tile (8 VGPRs)
  float lacc = 0.0f;       // scalar fallback accumulator
  const v2f aones = {1.0f, 1.0f};

  for (int bb = 0; bb < BPW; ++bb) {
    const int b     = bg * BG + wave * BPW + bb;
    const int gbase = (b * C_SZ + c0) * HW;

    // stage 16 classes x 196 floats (contiguous 12,544 B) into LDS
    for (int i = 0; i < (TILE_V4 + 31) / 32; ++i) {
      int t = i * 32 + lane;
      if (t < TILE_V4) {
        int gidx = gbase + t * 4;
        if (gidx + 4 > CAM_ELEMS) gidx = CAM_ELEMS - 4;  // clamp last class tile
        copy16_g2lds(src + gidx, lbuf + t * 4);
      }
    }
    wait_async0();

    // 49 x (ds_load_b64 -> v_wmma_f32_16x16x4_f32), K-depth 4 per step
    for (int bk = 0; bk < HW; bk += 4) {
      v2f bv = *(const v2f*)(bbase + bk);
#if HAVE_WMMA_F32
      acc = __builtin_amdgcn_wmma_f32_16x16x4_f32(
          /*neg_a=*/false, aones, /*neg_b=*/false, bv,
          /*c_mod=*/(short)0, acc, /*reuse_a=*/false, /*reuse_b=*/false);
#else
      lacc += bv.x + bv.y;
#endif
    }
  }

  float csum;
#if HAVE_WMMA_F32
  csum = acc[0];  // row M=0: lanes 0..15 hold class-column sums
#else
  csum = lacc + __shfl_down(lacc, 16, 32);
#endif
  const int c = c0 + ncls;
  if (lane < 16 && c < C_SZ) atomicAdd(&sums[cam * C_SZ + c], csum);
}

// ---- kernel 2: bincount + mean |avg_conf - avg_count| ---------------------
__global__ void mdca_finalize(const int* __restrict__ target,
                              const float* __restrict__ sums,
                              float* __restrict__ out) {
  __shared__ int counts[C_SZ];
  __shared__ float red[256];
  const int tid = threadIdx.x;  // blockDim = 256

  for (int c = tid; c < C_SZ; c += 256) counts[c] = 0;
  __syncthreads();
  for (int i = tid; i < B_SZ; i += 256) atomicAdd(&counts[target[i]], 1);
  __syncthreads();

  const float invBHW = 1.0f / (float)(B_SZ * HW);
  const float invB   = 1.0f / (float)B_SZ;
  float l[4] = {0.f, 0.f, 0.f, 0.f};
  for (int c = tid; c < C_SZ; c += 256) {
    const float cnt = (float)counts[c] * invB;
#pragma unroll
    for (int e = 0; e < 4; ++e)
      l[e] += fabsf(sums[e * C_SZ + c] * invBHW - cnt);
  }
#pragma unroll
  for (int e = 0; e < 4; ++e) {
    red[tid] = l[e];
    __syncthreads();
    for (int s = 128; s > 0; s >>= 1) {
      if (tid < s) red[tid] += red[tid + s];
      __syncthreads();
    }
    if (tid == 0) out[e] = red[0] * (1.0f / (float)C_SZ);
    __syncthreads();
  }
}

extern "C" void kernel_launch(void* const* d_in, const int* in_sizes, int n_in,
                              void* d_out, int out_size, void* d_ws, size_t ws_size,
                              hipStream_t stream) {
  (void)in_sizes; (void)n_in; (void)out_size; (void)ws_size;
  const float* c0 = (const float*)d_in[0];
  const float* c1 = (const float*)d_in[1];
  const float* c2 = (const float*)d_in[2];
  const float* c3 = (const float*)d_in[3];
  const int* target = (const int*)d_in[4];
  float* sums = (float*)d_ws;  // 4 * 1000 floats = 16 KB

  mdca_zero<<<(4 * C_SZ + 255) / 256, 256, 0, stream>>>(sums, 4 * C_SZ);

  const size_t smem = (size_t)WAVES * TILE_FLOATS * sizeof(float);  // 50,176 B
  mdca_reduce<<<dim3(4 * NCT * NBG), dim3(WAVES * 32), smem, stream>>>(
      c0, c1, c2, c3, sums);

  mdca_finalize<<<dim3(1), dim3(256), 0, stream>>>(target, sums, (float*)d_out);
}